// RNN_3848290697656
// MI455X (gfx1250) — compile-verified
//
#include <hip/hip_runtime.h>

typedef float v2f __attribute__((ext_vector_type(2)));
typedef float v8f __attribute__((ext_vector_type(8)));
typedef __attribute__((address_space(3))) char* lds_ptr_t;

#define DD     200      // feature dim
#define DP     208      // padded feature dim (13 tiles of 16)
#define SQ     512      // sequence length
#define NB     16       // batch
#define MROWS  (NB*SQ)  // 8192
#define NVOCAB 10000
#define LAYERS 3
#define KSTEPS 50       // K=200 / 4
#define FRAG   64       // floats per 16x16x4 fragment (32 lanes x v2f)

__device__ __forceinline__ v8f wmma_f32(v2f a, v2f b, v8f c) {
  // V_WMMA_F32_16X16X4_F32 : D = A(16x4) x B(4x16) + C, fp32 throughout
  return __builtin_amdgcn_wmma_f32_16x16x4_f32(
      /*neg_a=*/false, a, /*neg_b=*/false, b,
      /*c_mod=*/(short)0, c, /*reuse_a=*/false, /*reuse_b=*/false);
}

// ---------------------------------------------------------------------------
// 1) Embedding gather into padded h workspace (pad columns = 0.0 exactly)
// ---------------------------------------------------------------------------
__global__ void k_gather(const int* __restrict__ x, const float* __restrict__ emb,
                         float* __restrict__ h) {
  int idx = blockIdx.x * blockDim.x + threadIdx.x;
  if (idx >= MROWS * DP) return;
  int row = idx / DP;
  int col = idx - row * DP;
  float v = 0.0f;
  if (col < DD) v = emb[(size_t)x[row] * DD + col];
  h[idx] = v;
}

// ---------------------------------------------------------------------------
// 1b) Pad one layer's w_ih (200x200) into 208 rows (rows 200..207 = 0)
// ---------------------------------------------------------------------------
__global__ void k_padw(const float* __restrict__ w, float* __restrict__ wpad) {
  int idx = blockIdx.x * blockDim.x + threadIdx.x;
  if (idx >= DP * DD) return;
  int n = idx / DD;
  wpad[idx] = (n < DD) ? w[idx] : 0.0f;
}

// ---------------------------------------------------------------------------
// 1c) Pack a row-major matrix (rows x 200, row stride `stride`) into WMMA
//     fragment order: frag[tile][kk][lane*2+j] = src[(tile*16+lane&15)*stride
//     + kk*4 + 2*(lane>>4) + j].  Source reads are coalesced; every later
//     fragment load in the GEMM becomes one contiguous 256B wave transaction.
// ---------------------------------------------------------------------------
__global__ void k_pack(const float* __restrict__ src, float* __restrict__ dst,
                       int rows, int stride) {
  int idx = blockIdx.x * blockDim.x + threadIdx.x;          // over rows*200
  if (idx >= rows * DD) return;
  int r = idx / DD;
  int k = idx - r * DD;
  int tile = r >> 4, lr = r & 15;
  int kk = k >> 2, rem = k & 3;
  int half = rem >> 1, j = rem & 1;
  dst[((size_t)(tile * KSTEPS + kk) << 6) + ((half << 4) + lr) * 2 + j] =
      src[(size_t)r * stride + k];
}

// ---------------------------------------------------------------------------
// 2) xw = h @ w_ih^T + (b_ih + b_hh); wpad zero-padded to 208 rows.
// ---------------------------------------------------------------------------
__global__ void k_xw(const float* __restrict__ h, const float* __restrict__ wpad,
                     const float* __restrict__ bih, const float* __restrict__ bhh,
                     float* __restrict__ xw) {
  const int lane = threadIdx.x & 31;
  const int half = lane >> 4;
  const int lr   = lane & 15;
  const int wave = (blockIdx.x * blockDim.x + threadIdx.x) >> 5;
  const int mt = wave / 13;
  const int nt = wave - mt * 13;
  const int m0 = mt * 16, n0 = nt * 16;
  const int n  = n0 + lr;

  v8f c = {};
  const float* arow = h    + (size_t)(m0 + lr) * DP + 2 * half;
  const float* brow = wpad + (size_t)n * DD        + 2 * half;
  for (int k0 = 0; k0 < DD; k0 += 4) {
    v2f a = *(const v2f*)(arow + k0);
    v2f b = *(const v2f*)(brow + k0);
    c = wmma_f32(a, b, c);
  }
  float bias = (n < DD) ? (bih[n] + bhh[n]) : 0.0f;
#pragma unroll
  for (int j = 0; j < 8; ++j) {
    int m = m0 + j + 8 * half;
    xw[(size_t)m * DP + n] = c[j] + bias;     // pad cols store exact 0
  }
}

// ---------------------------------------------------------------------------
// 3) Sequential scan: h_t = tanh(xw_t + h_{t-1} @ w_hh^T)
//    One workgroup, 13 waves. W_hh slice in 100 VGPRs for all 512 steps.
// ---------------------------------------------------------------------------
__global__ void k_rnn(const float* __restrict__ xw, const float* __restrict__ whh,
                      float* __restrict__ h, float* __restrict__ hid_out) {
  __shared__ float sh[NB * DP];
  const int lane = threadIdx.x & 31;
  const int half = lane >> 4;
  const int lr   = lane & 15;
  const int wv   = threadIdx.x >> 5;
  const int n0   = wv * 16;
  const int n    = n0 + lr;
  const int ncl  = (n < DD) ? n : 0;

  v2f breg[KSTEPS];
#pragma unroll
  for (int kk = 0; kk < KSTEPS; ++kk) {
    v2f b = *(const v2f*)(whh + (size_t)ncl * DD + kk * 4 + 2 * half);
    if (n >= DD) { b[0] = 0.0f; b[1] = 0.0f; }
    breg[kk] = b;
  }

  for (int i = threadIdx.x; i < NB * DP; i += blockDim.x) sh[i] = 0.0f;
  __syncthreads();

  for (int t = 0; t < SQ; ++t) {
    v8f c;
#pragma unroll
    for (int j = 0; j < 8; ++j) {
      int b = j + 8 * half;
      c[j] = xw[((size_t)b * SQ + t) * DP + n];
    }
#pragma unroll
    for (int kk = 0; kk < KSTEPS; ++kk) {
      v2f a = *(const v2f*)(&sh[lr * DP + kk * 4 + 2 * half]);
      c = wmma_f32(a, breg[kk], c);
    }
    float hn[8];
#pragma unroll
    for (int j = 0; j < 8; ++j) hn[j] = tanhf(c[j]);

    __syncthreads();
#pragma unroll
    for (int j = 0; j < 8; ++j) {
      int b = j + 8 * half;
      sh[b * DP + n] = hn[j];
      h[((size_t)b * SQ + t) * DP + n] = hn[j];
    }
    if (t == SQ - 1 && n < DD) {
#pragma unroll
      for (int j = 0; j < 8; ++j) {
        int b = j + 8 * half;
        hid_out[(size_t)b * DD + n] = hn[j];
      }
    }
    __syncthreads();
  }
}

// ---------------------------------------------------------------------------
// 4) logits = h @ fc_w^T + fc_b ; 32.8 GFLOP, fp32 WMMA.
//    Block = 8 waves sharing 5 N-tiles; B fragments (64 KB) staged in LDS via
//    gfx1250 global_load_async_to_lds_b128 (+ s_wait_asynccnt). Each wave:
//    4 M-tiles x 5 N-tiles, A from fragment-packed global (coalesced 256B),
//    B from LDS. 20 WMMAs per {4 global b64 + 5 ds b64}. Logits stored
//    non-temporally (328 MB stream > 192 MB L2) so fc operands stay resident.
// ---------------------------------------------------------------------------
__global__ void k_fc(const float* __restrict__ afrag, const float* __restrict__ bfrag,
                     const float* __restrict__ fcb, float* __restrict__ out) {
  extern __shared__ char smem[];                 // 5*50*256 = 64000 bytes
  const int lane = threadIdx.x & 31;
  const int half = lane >> 4;
  const int lr   = lane & 15;
  const int wave = threadIdx.x >> 5;             // 0..7
  const int bm = blockIdx.x / 125;               // 0..15 -> 32 M-tiles/block
  const int nq = blockIdx.x - bm * 125;          // 0..124 -> 5 N-tiles
  const int mbase = bm * 32 + wave * 4;          // first M-tile of this wave
  const int n0 = nq * 80;

  // --- async stage of the block's B fragments into LDS --------------------
  {
    const char* gcol = (const char*)(bfrag + (size_t)(nq * 5) * (KSTEPS * FRAG));
    for (int i = threadIdx.x; i < 4000; i += 256) {   // 4000 x 16B = 64000B
      asm volatile("global_load_async_to_lds_b128 %0, %1, off"
                   :: "v"((lds_ptr_t)(smem + i * 16)),
                      "v"(gcol + (size_t)i * 16)
                   : "memory");
    }
    asm volatile("s_wait_asynccnt 0" ::: "memory");
  }
  __syncthreads();
  const float* sB = (const float*)smem;

  v8f c[4][5] = {};
  const float* abase = afrag + ((size_t)mbase * KSTEPS << 6) + lane * 2;

  for (int kk = 0; kk < KSTEPS; ++kk) {
    v2f a[4];
#pragma unroll
    for (int mi = 0; mi < 4; ++mi)
      a[mi] = *(const v2f*)(abase + (((size_t)mi * KSTEPS + kk) << 6));
#pragma unroll
    for (int tj = 0; tj < 5; ++tj) {
      v2f b = *(const v2f*)(&sB[((tj * KSTEPS + kk) << 6) + lane * 2]);
#pragma unroll
      for (int mi = 0; mi < 4; ++mi)
        c[mi][tj] = wmma_f32(a[mi], b, c[mi][tj]);
    }
  }

#pragma unroll
  for (int tj = 0; tj < 5; ++tj) {
    int nn = n0 + tj * 16 + lr;
    float bias = fcb[nn];
#pragma unroll
    for (int mi = 0; mi < 4; ++mi) {
#pragma unroll
      for (int j = 0; j < 8; ++j) {
        float* p = out + (size_t)((mbase + mi) * 16 + j + 8 * half) * NVOCAB + nn;
        __builtin_nontemporal_store(c[mi][tj][j] + bias, p);
      }
    }
  }
}

// ---------------------------------------------------------------------------
extern "C" void kernel_launch(void* const* d_in, const int* in_sizes, int n_in,
                              void* d_out, int out_size, void* d_ws, size_t ws_size,
                              hipStream_t stream) {
  (void)in_sizes; (void)n_in; (void)out_size; (void)ws_size;
  const int*   x   = (const int*)  d_in[0];
  const float* emb = (const float*)d_in[1];
  const float* wih = (const float*)d_in[2];
  const float* whh = (const float*)d_in[3];
  const float* bih = (const float*)d_in[4];
  const float* bhh = (const float*)d_in[5];
  const float* fcw = (const float*)d_in[6];
  const float* fcb = (const float*)d_in[7];

  float* logits = (float*)d_out;                       // (16,512,10000)
  float* hid    = logits + (size_t)MROWS * NVOCAB;     // (3,16,200)

  float* h_ws  = (float*)d_ws;                         // (8192,208)
  float* xw_ws = h_ws  + (size_t)MROWS * DP;           // (8192,208)
  float* w_pad = xw_ws + (size_t)MROWS * DP;           // (208,200)
  float* bfrag = w_pad + (size_t)DP * DD;              // 625*50*64 = 2.0M f
  float* afrag = xw_ws;                                // alias: xw dead by then

  k_gather<<<(MROWS * DP + 255) / 256, 256, 0, stream>>>(x, emb, h_ws);

  for (int l = 0; l < LAYERS; ++l) {
    k_padw<<<(DP * DD + 255) / 256, 256, 0, stream>>>(wih + (size_t)l * DD * DD,
                                                      w_pad);
    k_xw<<<(512 * 13) / 8, 256, 0, stream>>>(h_ws, w_pad,
                                             bih + (size_t)l * DD,
                                             bhh + (size_t)l * DD, xw_ws);
    k_rnn<<<1, 416, 0, stream>>>(xw_ws, whh + (size_t)l * DD * DD, h_ws,
                                 hid + (size_t)l * NB * DD);
  }

  // Fragment-pack both FC operands (coalesced one-time passes).
  k_pack<<<(NVOCAB * DD + 255) / 256, 256, 0, stream>>>(fcw, bfrag, NVOCAB, DD);
  k_pack<<<(MROWS * DD + 255) / 256, 256, 0, stream>>>(h_ws, afrag, MROWS, DP);

  k_fc<<<16 * 125, 256, 64000, stream>>>(afrag, bfrag, fcb, logits);
}